// DTCWTLayerQshift_bp_30253749633566
// MI455X (gfx1250) — compile-verified
//
#include <hip/hip_runtime.h>

typedef __attribute__((ext_vector_type(2))) float v2f;
typedef __attribute__((ext_vector_type(8))) float v8f;
typedef __attribute__((ext_vector_type(4))) unsigned int v4u;
typedef __attribute__((ext_vector_type(8))) int v8i;
typedef __attribute__((ext_vector_type(4))) int v4i;

#define NB 16
#define NC 64
#define NH 256
#define NW 256

// ---------------- LDS layout (units: floats) ----------------
#define F_OFF     0
#define F_SIZE    (3 * 16 * 56)            // 3 filter banks, 16x56 banded matrices
#define XE_OFF    (F_OFF + F_SIZE)         // 2688
#define XE_W      284                      // 256 + 28 reflected halo
#define XE_ROWS   96                       // 88 needed + pad to 6x16 tiles
#define XE_SIZE   (XE_ROWS * XE_W)         // 27264
#define RF_OFF    (XE_OFF + XE_SIZE)       // 29952
#define RF_STRIDE 132                      // 128 cols + 4 pad (bank spread)
#define RF_ROWS   96
#define RF_SIZE   (3 * RF_ROWS * RF_STRIDE) // 38016
#define SMEM_FLOATS (RF_OFF + RF_SIZE)     // 67968 floats = 271,872 B < 320 KB
// high-band scratch reuses the Xext region after stage-1 barrier
#define HI_OFF    XE_OFF
#define HI_STRIDE 132

__device__ __forceinline__ int refl256(int t) {
  t = (t < 0) ? (-1 - t) : t;
  return (t > 255) ? (511 - t) : t;
}

__global__ void __launch_bounds__(256, 1)
dtcwt_fused(const float* __restrict__ X,
            const float* __restrict__ h0a, const float* __restrict__ h0b,
            const float* __restrict__ h1a, const float* __restrict__ h1b,
            const float* __restrict__ h2a, const float* __restrict__ h2b,
            float* __restrict__ out)
{
  extern __shared__ float smem[];
  const int tid  = threadIdx.x;
  const int lane = tid & 31;
  const int wave = tid >> 5;
  const int band = blockIdx.x & 3;    // 4 bands of 32 coldfilt output rows
  const int bc   = blockIdx.x >> 2;   // 0..1023 (b*64 + c)

  const int half = (lane < 16) ? 0 : 1;  // which 16-lane half of the wave
  const int ln   = lane & 15;
  const int koff = half * 2;             // K offset within a K=4 WMMA step

  const int rowbase = 64 * band - 12;    // first source row (pre-reflection)
  const float* Xs = X + (long long)bc * (NH * NW);

  // Valid (non-reflected) source-row window: the TDM moves exactly these rows.
  const int r0    = rowbase < 0 ? 0 : rowbase;
  const int r1v   = (rowbase + 95 > 255) ? 255 : (rowbase + 95);
  const int nrows = r1v - r0 + 1;
  const int rr_lo = r0 - rowbase;        // first Xext row written by the TDM
  const int rr_hi = rr_lo + nrows;       // one past last

  // ---- phase 0a: issue Tensor-DMA for the interior of the X tile ----
  // 2D tile: nrows x 256 f32, row stride 256; LDS destination gets 28 DWORDs
  // of padding after every 256 DWORDs => hardware reproduces the 284-float
  // Xext row stride. Destination starts at column 12 of row rr_lo.
  if (wave == 0) {
    unsigned long long gaddr =
        (unsigned long long)(const void*)(Xs + (long long)r0 * NW);
    unsigned int lds_byte =
        (unsigned int)(unsigned long long)(const void*)
            (&smem[XE_OFF + rr_lo * XE_W + 12]);   // LDS aperture: addr[31:0]
    v4u g0;
    g0[0] = 1u;                                       // count=1, user mode
    g0[1] = lds_byte;                                 // lds_addr (bytes)
    g0[2] = (unsigned int)(gaddr & 0xffffffffull);    // global_addr[31:0]
    g0[3] = (unsigned int)((gaddr >> 32) & 0x1ffffffull) | (2u << 30); // type=2
    v8i g1;
    g1[0] = (2 << 16)      // data_size = 4 bytes
          | (1 << 20)      // pad_enable
          | (7 << 22)      // pad_interval code 7 = 256 DWORDs
          | (27 << 25);    // pad_amount  code 27 = 28 DWORDs
    g1[1] = (int)(256u << 16);                  // tensor_dim0 = 256 (low 16 b)
    g1[2] = (int)(((unsigned)nrows) << 16);     // tensor_dim1 low 16 bits
    g1[3] = (int)(256u << 16);                  // tile_dim0 = 256
    g1[4] = nrows;                              // tile_dim1 = nrows
    g1[5] = 256;                                // tensor_dim0_stride = 256
    g1[6] = 0;
    g1[7] = 0;
    v4i gz = {0, 0, 0, 0};
#if defined(__clang_major__) && (__clang_major__ >= 23)
    v8i gz8 = {0, 0, 0, 0, 0, 0, 0, 0};
    __builtin_amdgcn_tensor_load_to_lds(g0, g1, gz, gz, gz8, 0);
#else
    __builtin_amdgcn_tensor_load_to_lds(g0, g1, gz, gz, 0);
#endif
  }

  // ---- phase 0b: zero filter matrices + fill reflected halo (runs during DMA)
  for (int i = tid; i < F_SIZE; i += 256) smem[F_OFF + i] = 0.0f;
  for (int i = tid; i < XE_ROWS * XE_W; i += 256) {
    int rr = i / XE_W;
    int e  = i - rr * XE_W;
    bool covered = (rr >= rr_lo) && (rr < rr_hi) && (e >= 12) && (e < 268);
    if (!covered) {
      int sr = refl256(rowbase + rr);
      int sc = refl256(e - 12);
      smem[XE_OFF + i] = Xs[sr * NW + sc];
    }
  }
  if (wave == 0) __builtin_amdgcn_s_wait_tensorcnt(0);
  __syncthreads();

  // ---- phase 0c: fill banded polyphase filter matrices F[bank][j][p] ----
  // out[2t]   uses taps at p = 4t+2i   (even phase)  [lowpass: ha]
  // out[2t+1] uses taps at p = 4t+2i+1 (odd  phase)  [lowpass: hb]
  // highpass banks swap y1/y2.
  for (int i = tid; i < 3 * 8 * 14; i += 256) {
    int f = i / (8 * 14);
    int r = i - f * (8 * 14);
    int t = r / 14;
    int k = r - t * 14;
    int pe = 4 * t + 2 * k;       // even-phase tap position
    if (f == 0) {                 // (ha,hb) = (h0b,h0a), lowpass
      smem[F_OFF + (0 * 16 + 2 * t    ) * 56 + pe    ] = h0b[k];
      smem[F_OFF + (0 * 16 + 2 * t + 1) * 56 + pe + 1] = h0a[k];
    } else if (f == 1) {          // (ha,hb) = (h1b,h1a), highpass
      smem[F_OFF + (1 * 16 + 2 * t    ) * 56 + pe + 1] = h1a[k];
      smem[F_OFF + (1 * 16 + 2 * t + 1) * 56 + pe    ] = h1b[k];
    } else {                      // (ha,hb) = (h2b,h2a), highpass
      smem[F_OFF + (2 * 16 + 2 * t    ) * 56 + pe + 1] = h2a[k];
      smem[F_OFF + (2 * 16 + 2 * t + 1) * 56 + pe    ] = h2b[k];
    }
  }
  __syncthreads();

  // ---- stage 1: row filtering via WMMA. 6 rowgroups x 8 wgroups x 3 banks ----
  // D[m=w_out(16), n=row(16)] = sum_p F[m,p] * Xext[row, e0+p], 14 K=4 steps.
  for (int tile = wave; tile < 144; tile += 8) {
    int f  = tile % 3;
    int r2 = tile / 3;
    int wg = r2 & 7;
    int rg = r2 >> 3;
    const int e0  = 32 * wg;
    const int rrb = 16 * rg;
    const float* Fm = &smem[F_OFF + (f * 16 + ln) * 56];          // A: lane=M row
    const float* Bx = &smem[XE_OFF + (rrb + ln) * XE_W + e0];     // B: lane=N col
    v8f acc = {};
#pragma unroll
    for (int kk = 0; kk < 14; ++kk) {
      v2f a, b;
      a.x = Fm[4 * kk + koff];     a.y = Fm[4 * kk + koff + 1];
      b.x = Bx[4 * kk + koff];     b.y = Bx[4 * kk + koff + 1];
      acc = __builtin_amdgcn_wmma_f32_16x16x4_f32(
          false, a, false, b, (short)0, acc, false, false);
    }
    // D: lane n holds N=n (row), VGPR r holds M=r (+8 for high lanes) (w_out)
    float* dst = &smem[RF_OFF + (f * RF_ROWS + rrb + ln) * RF_STRIDE
                       + 16 * wg + 8 * half];
#pragma unroll
    for (int r = 0; r < 8; ++r) dst[r] = acc[r];
  }
  __syncthreads();

  // ---- stage 2: column filtering via WMMA. 4 outputs x 2 h-groups x 8 w-groups
  // D[m=h_out(16), n=w(16)] = sum_p F[m,p] * RF[rr0+p, w0+n]
  {
    const int srcs[4]  = {0, 0, 1, 2};   // Lo, Lo, Hi, Mid
    const int banks[4] = {0, 1, 0, 2};   // h0-low, h1-high, h0-low, h2-high
    for (int tile = wave; tile < 64; tile += 8) {
      int o  = tile & 3;                 // 0:LoLo 1:LoHi 2:HiLo 3:MidMid
      int r2 = tile >> 2;
      int wg = r2 & 7;
      int g2 = r2 >> 3;
      const int src = srcs[o];
      const int f   = banks[o];
      const int rr0 = 32 * g2;
      const int w0  = 16 * wg;
      const float* Fm = &smem[F_OFF + (f * 16 + ln) * 56];
      const float* Bc = &smem[RF_OFF + (src * RF_ROWS + rr0) * RF_STRIDE + w0 + ln];
      v8f acc = {};
#pragma unroll
      for (int kk = 0; kk < 14; ++kk) {
        v2f a, b;
        a.x = Fm[4 * kk + koff];               a.y = Fm[4 * kk + koff + 1];
        b.x = Bc[(4 * kk + koff) * RF_STRIDE]; b.y = Bc[(4 * kk + koff + 1) * RF_STRIDE];
        acc = __builtin_amdgcn_wmma_f32_16x16x4_f32(
            false, a, false, b, (short)0, acc, false, false);
      }
      if (o == 0) {
        // LoLo (16,64,128,128) -> d_out[0 ..)
        const int hb = band * 32 + g2 * 16 + 8 * half;
        float* dst = out + ((long long)bc * 128 + hb) * 128 + w0 + ln;
#pragma unroll
        for (int r = 0; r < 8; ++r) dst[r * 128] = acc[r];
      } else {
        float* dst = &smem[HI_OFF + ((o - 1) * 32 + g2 * 16 + 8 * half) * HI_STRIDE
                           + w0 + ln];
#pragma unroll
        for (int r = 0; r < 8; ++r) dst[r * HI_STRIDE] = acc[r];
      }
    }
  }
  __syncthreads();

  // ---- stage 3: q2c + stores of Yhr/Yhi ----
  {
    const float sc = 0.70710678118654752f;     // sqrt(0.5)
    const int i1[3] = {0, 1, 2};               // LoHi->15, HiLo->45, MidMid->75
    const int i2[3] = {5, 4, 3};               // LoHi->165, HiLo->135, MidMid->105
    const long long YR = 16LL * 64 * 128 * 128;          // 16,777,216
    const long long YI = YR + 16LL * 64 * 6 * 64 * 64;   // 41,943,040
    const int HQ0 = band * 16;
    for (int i = tid; i < 3 * 16 * 64; i += 256) {
      int s  = i >> 10;            // source 0..2
      int r  = i & 1023;
      int hq = r >> 6;
      int wq = r & 63;
      const float* Sb = &smem[HI_OFF + (s * 32 + 2 * hq) * HI_STRIDE + 2 * wq];
      float a = Sb[0];
      float b = Sb[1];
      float c = Sb[HI_STRIDE];
      float d = Sb[HI_STRIDE + 1];
      long long p1 = ((long long)(bc * 6 + i1[s]) * 64 + HQ0 + hq) * 64 + wq;
      long long p2 = ((long long)(bc * 6 + i2[s]) * 64 + HQ0 + hq) * 64 + wq;
      out[YR + p1] = (a - d) * sc;
      out[YI + p1] = (b + c) * sc;
      out[YR + p2] = (a + d) * sc;
      out[YI + p2] = (b - c) * sc;
    }
  }
}

extern "C" void kernel_launch(void* const* d_in, const int* in_sizes, int n_in,
                              void* d_out, int out_size, void* d_ws, size_t ws_size,
                              hipStream_t stream) {
  (void)in_sizes; (void)n_in; (void)out_size; (void)d_ws; (void)ws_size;
  const float* X   = (const float*)d_in[0];
  const float* h0a = (const float*)d_in[1];
  const float* h0b = (const float*)d_in[2];
  const float* h1a = (const float*)d_in[3];
  const float* h1b = (const float*)d_in[4];
  const float* h2a = (const float*)d_in[5];
  const float* h2b = (const float*)d_in[6];

  const size_t shmem = (size_t)SMEM_FLOATS * sizeof(float);  // 271,872 B
  hipFuncSetAttribute(reinterpret_cast<const void*>(dtcwt_fused),
                      hipFuncAttributeMaxDynamicSharedMemorySize, (int)shmem);

  dim3 grid(NB * NC * 4);   // 4096 workgroups: (b,c) x 4 H-bands
  dtcwt_fused<<<grid, 256, shmem, stream>>>(X, h0a, h0b, h1a, h1b, h2a, h2b,
                                            (float*)d_out);
}